// ScatteredExperts_66271345377806
// MI455X (gfx1250) — compile-verified
//
#include <hip/hip_runtime.h>

typedef float v2f __attribute__((ext_vector_type(2)));
typedef float v8f __attribute__((ext_vector_type(8)));

#define T_TOK   4096
#define NEXP    8
#define KFAN    2
#define DIN     1024
#define DOUT    1024
#define NROWS   (T_TOK * KFAN)      // 8192 scattered rows

#define TILE_M  32                  // scattered rows per block (2 x 16 M-tiles)
#define WAVE_N  64                  // output cols per wave (4 x 16 subtiles)
#define NWAVES  4
#define BLOCK_N (WAVE_N * NWAVES)   // 256 output cols per block
#define KCHUNK  256                 // K-dim staged in LDS per iteration
#define LDS_PAD 4
#define LDS_STR (KCHUNK + LDS_PAD)  // 260 floats -> conflict-free A-frag reads

static __device__ __forceinline__ v8f wmma_f32(v2f a, v2f b, v8f c) {
    return __builtin_amdgcn_wmma_f32_16x16x4_f32(
        false, a, false, b, (short)0, c, false, false);
}

// One KCHUNK of the 32xWAVE_N GEMM for this wave: each B fragment feeds two
// 16-row M-tiles (rows m and m+16), doubling wmma per byte of B traffic.
// MASKED=false: whole 32-row tile owned by one expert (common case).
// MASKED=true : boundary tile; rows not owned by expert e zeroed via mf0/mf1.
template<bool MASKED>
static __device__ __forceinline__ void gemm_chunk(
    const float* __restrict__ sArow0,  // &sA[m   ][0] for this lane
    const float* __restrict__ sArow1,  // &sA[m+16][0] for this lane
    int kb,                            // K sub-offset (0 or 2)
    const float* __restrict__ wcol,    // &W[e][kc][colBase]
    float mf0, float mf1,
    v8f& c00, v8f& c01, v8f& c02, v8f& c03,   // M-tile 0, subtiles 0..3
    v8f& c10, v8f& c11, v8f& c12, v8f& c13)   // M-tile 1, subtiles 0..3
{
    #pragma unroll 2
    for (int k0 = 0; k0 < KCHUNK; k0 += 4) {
        v2f a0 = *(const v2f*)(sArow0 + k0 + kb);   // ds_load_b64, always exec
        v2f a1 = *(const v2f*)(sArow1 + k0 + kb);
        if (MASKED) { a0 *= mf0; a1 *= mf1; }

        const float* wrow = wcol + (size_t)(k0 + kb) * DOUT;
        v2f b0 = { wrow[0],  wrow[DOUT]      };     // W[k][n], W[k+1][n]
        v2f b1 = { wrow[16], wrow[DOUT + 16] };
        v2f b2 = { wrow[32], wrow[DOUT + 32] };
        v2f b3 = { wrow[48], wrow[DOUT + 48] };

        c00 = wmma_f32(a0, b0, c00);
        c10 = wmma_f32(a1, b0, c10);
        c01 = wmma_f32(a0, b1, c01);
        c11 = wmma_f32(a1, b1, c11);
        c02 = wmma_f32(a0, b2, c02);
        c12 = wmma_f32(a1, b2, c12);
        c03 = wmma_f32(a0, b3, c03);
        c13 = wmma_f32(a1, b3, c13);
    }
}

__global__ __launch_bounds__(128)
void scattered_experts_wmma_f32(const float* __restrict__ x,
                                const float* __restrict__ weight,
                                const int*   __restrict__ bin_ids,
                                const int*   __restrict__ indices,
                                const float* __restrict__ gates,
                                float*       __restrict__ out)
{
    __shared__ float sA[TILE_M][LDS_STR];
    __shared__ int   sBin[TILE_M];
    __shared__ int   sTok[TILE_M];
    __shared__ float sGate[TILE_M];

    const int r0      = blockIdx.x * TILE_M;   // first scattered row of tile
    const int n_block = blockIdx.y * BLOCK_N;  // first output column of block
    const int tid     = threadIdx.x;
    const int wave    = tid >> 5;
    const int lane    = tid & 31;

    if (tid < TILE_M) {
        const int r   = r0 + tid;
        const int idx = indices[r];
        sBin[tid]  = bin_ids[r];
        sTok[tid]  = idx / KFAN;   // source token for this scattered row
        sGate[tid] = gates[idx];   // gate for this scattered row
    }
    __syncthreads();

    // A-fragment (16x4 f32): lanes 0-15 hold K=0,1; lanes 16-31 hold K=2,3
    const int m0 = lane & 15;            // A row for M-tile 0
    const int m1 = m0 + 16;              // A row for M-tile 1
    const int kb = (lane >> 4) * 2;      // K sub-offset (0 or 2)
    const int binRow0 = sBin[m0];
    const int binRow1 = sBin[m1];
    const int eFirst  = sBin[0];
    const int eLast   = sBin[TILE_M - 1]; // bin_ids sorted -> contiguous span

    const int colBase = n_block + wave * WAVE_N + (lane & 15);

    v8f c00 = {}, c01 = {}, c02 = {}, c03 = {};
    v8f c10 = {}, c11 = {}, c12 = {}, c13 = {};

    for (int kc = 0; kc < DIN; kc += KCHUNK) {
        // Cooperative gather of A tile: 32 rows x KCHUNK floats from x[token]
        #pragma unroll
        for (int i = 0; i < (TILE_M * KCHUNK) / (4 * 128); ++i) {
            const int f    = tid + i * 128;        // float4 index 0..2047
            const int row  = f >> 6;               // 64 float4 per row
            const int col4 = f & 63;
            const float4 v =
                ((const float4*)(x + (size_t)sTok[row] * DIN + kc))[col4];
            *(float4*)&sA[row][col4 * 4] = v;
        }
        __syncthreads();

        if (eFirst == eLast) {
            // Fast path: whole tile owned by one expert, no masking.
            const float* wcol = weight + (size_t)eFirst * DIN * DOUT
                                       + (size_t)kc * DOUT + colBase;
            gemm_chunk<false>(&sA[m0][0], &sA[m1][0], kb, wcol, 1.0f, 1.0f,
                              c00, c01, c02, c03, c10, c11, c12, c13);
        } else {
            // Boundary tile (rare): accumulate each expert with rows masked.
            for (int e = eFirst; e <= eLast; ++e) {
                const float mf0 = (binRow0 == e) ? 1.0f : 0.0f;
                const float mf1 = (binRow1 == e) ? 1.0f : 0.0f;
                const float* wcol = weight + (size_t)e * DIN * DOUT
                                           + (size_t)kc * DOUT + colBase;
                gemm_chunk<true>(&sA[m0][0], &sA[m1][0], kb, wcol, mf0, mf1,
                                 c00, c01, c02, c03, c10, c11, c12, c13);
            }
        }
        __syncthreads();
    }

    // Epilogue: gated scatter-add back to token order.
    // C/D layout: VGPR v, lanes 0-15 -> M=v, lanes 16-31 -> M=v+8, N=lane&15.
    const int mBase = (lane >> 4) * 8;
    #pragma unroll
    for (int v = 0; v < 8; ++v) {
        const int row0 = mBase + v;            // row in M-tile 0 (tile rows 0..15)
        const int row1 = row0 + 16;            // row in M-tile 1 (tile rows 16..31)
        {
            const float g = sGate[row0];
            float* o = out + (size_t)sTok[row0] * DOUT + colBase;
            atomicAdd(o,      c00[v] * g);
            atomicAdd(o + 16, c01[v] * g);
            atomicAdd(o + 32, c02[v] * g);
            atomicAdd(o + 48, c03[v] * g);
        }
        {
            const float g = sGate[row1];
            float* o = out + (size_t)sTok[row1] * DOUT + colBase;
            atomicAdd(o,      c10[v] * g);
            atomicAdd(o + 16, c11[v] * g);
            atomicAdd(o + 32, c12[v] * g);
            atomicAdd(o + 48, c13[v] * g);
        }
    }
}

extern "C" void kernel_launch(void* const* d_in, const int* in_sizes, int n_in,
                              void* d_out, int out_size, void* d_ws, size_t ws_size,
                              hipStream_t stream) {
    const float* x        = (const float*)d_in[0];
    const float* weight   = (const float*)d_in[1];
    const int*   bin_ids  = (const int*)d_in[2];
    const int*   indices  = (const int*)d_in[3];
    // d_in[4] = padded_block_idxs (inert), d_in[5] = expert_offsets (unused:
    // bin_ids is sorted, so per-tile expert spans come from bin_ids directly)
    const float* gates    = (const float*)d_in[6];
    float*       out      = (float*)d_out;

    // Output is accumulated with atomics -> must start from zero every call.
    hipMemsetAsync(out, 0, (size_t)T_TOK * DOUT * sizeof(float), stream);

    dim3 grid(NROWS / TILE_M, DOUT / BLOCK_N);   // 256 x 4
    scattered_experts_wmma_f32<<<grid, 128, 0, stream>>>(
        x, weight, bin_ids, indices, gates, out);
}